// NMS_27831388078727
// MI455X (gfx1250) — compile-verified
//
#include <hip/hip_runtime.h>
#include <hip/hip_bf16.h>
#include <math.h>

typedef __attribute__((ext_vector_type(2))) float v2f;
typedef __attribute__((ext_vector_type(8))) float v8f;

#define NB   16384   // batch
#define SDIM 64      // state dim
#define WID  256     // hidden width

#define XS_STRIDE 36   // 64 x 32 A-tile, padded: A-frag gathers hit distinct banks
#define PK_STRIDE 160  // k-pair-interleaved B tile: row stride in floats (=32 mod 64 banks)

// ---------------------------------------------------------------------------
// OUT[M,N] = epilogue( X[M,K] @ B[K,N] )
//   TRANSB=0: B[k,n] = W[k*ldw + n]       TRANSB=1: B[k,n] = W[n*ldw + k]
//   MODE 0: raw
//   MODE 1: tanh(x + bias[n])
//   MODE 2: t = tanh(x + bias[n]); (1 - t*t) * aux[n]          (aux = w2 row)
//   MODE 3: x * (1 - h*h), h = aux[m*N + n]                    (aux = H0)
// fp32 WMMA 16x16x4; block tile 64x64 (8 waves, each 16x32), K-step 32.
// B tile stored k-pair interleaved: Bs[k/2][n] = {B[k,n], B[k+1,n]} so each
// B fragment is one aligned ds_load_b64 into an adjacent VGPR pair (no movs).
// ---------------------------------------------------------------------------
template <int TRANSB, int MODE>
__global__ void __launch_bounds__(256)
gemm_f32_wmma(const float* __restrict__ X, const float* __restrict__ W,
              int ldw, const float* __restrict__ bias,
              const float* __restrict__ aux,
              float* __restrict__ OUT, int M, int N, int K) {
  __shared__ __align__(16) float Xs[64 * XS_STRIDE];
  __shared__ __align__(16) float Bs[16 * PK_STRIDE];

  const int tid  = threadIdx.x;
  const int lane = tid & 31;
  const int wave = tid >> 5;
  const int wm   = wave & 3;        // M sub-tile (16 rows)
  const int wn   = wave >> 2;       // N sub-tile (32 cols)
  const int half = lane >> 4;       // lane half -> K pair select
  const int lr   = lane & 15;
  const int m0   = blockIdx.x * 64;
  const int n0   = blockIdx.y * 64;

  v8f acc0 = {0.f, 0.f, 0.f, 0.f, 0.f, 0.f, 0.f, 0.f};
  v8f acc1 = acc0;

  for (int k0 = 0; k0 < K; k0 += 32) {
    // ---- stage X tile (64 x 32), coalesced b128 ----
    {
      int r = tid >> 3;               // 0..31
      int c = (tid & 7) << 2;         // 0..28
      float4 x0 = *(const float4*)(X + (size_t)(m0 + r) * K + k0 + c);
      float4 x1 = *(const float4*)(X + (size_t)(m0 + r + 32) * K + k0 + c);
      *(float4*)(Xs + r * XS_STRIDE + c) = x0;
      *(float4*)(Xs + (r + 32) * XS_STRIDE + c) = x1;
    }
    // ---- stage B tile (32 x 64), k-pair interleaved ----
    if (TRANSB == 0) {
      int k2 = tid >> 4;              // 0..15 (pair of rows 2k2, 2k2+1)
      int n  = (tid & 15) << 2;       // 0..60
      float4 r0 = make_float4(0.f, 0.f, 0.f, 0.f), r1 = r0;
      if (n0 + n < N) {
        r0 = *(const float4*)(W + (size_t)(k0 + 2 * k2) * ldw + n0 + n);
        r1 = *(const float4*)(W + (size_t)(k0 + 2 * k2 + 1) * ldw + n0 + n);
      }
      float* p = Bs + k2 * PK_STRIDE + n * 2;
      *(float4*)(p)     = make_float4(r0.x, r1.x, r0.y, r1.y);
      *(float4*)(p + 4) = make_float4(r0.z, r1.z, r0.w, r1.w);
    } else {
      int n = tid >> 3;               // 0..31
      int k = (tid & 7) << 2;         // 0..28 (covers pairs k/2, k/2+1)
      float4 b0 = make_float4(0.f, 0.f, 0.f, 0.f), b1 = b0;
      if (n0 + n < N)       b0 = *(const float4*)(W + (size_t)(n0 + n) * ldw + k0 + k);
      if (n0 + n + 32 < N)  b1 = *(const float4*)(W + (size_t)(n0 + n + 32) * ldw + k0 + k);
      int k2 = k >> 1;
      *(v2f*)(Bs + k2 * PK_STRIDE + n * 2)              = (v2f){b0.x, b0.y};
      *(v2f*)(Bs + (k2 + 1) * PK_STRIDE + n * 2)        = (v2f){b0.z, b0.w};
      *(v2f*)(Bs + k2 * PK_STRIDE + (n + 32) * 2)       = (v2f){b1.x, b1.y};
      *(v2f*)(Bs + (k2 + 1) * PK_STRIDE + (n + 32) * 2) = (v2f){b1.z, b1.w};
    }
    __syncthreads();

    const int arow  = wm * 16 + lr;
    const int ncol2 = (wn * 32 + lr) * 2;
#pragma unroll
    for (int k4 = 0; k4 < 32; k4 += 4) {
      const int kb = k4 + 2 * half;   // A/B: lanes0-15 use K=k4,k4+1; lanes16-31 K=k4+2,k4+3
      const int k2 = kb >> 1;
      v2f a;
      a.x = Xs[arow * XS_STRIDE + kb];
      a.y = Xs[arow * XS_STRIDE + kb + 1];
      v2f b0v = *(const v2f*)(Bs + k2 * PK_STRIDE + ncol2);
      v2f b1v = *(const v2f*)(Bs + k2 * PK_STRIDE + ncol2 + 32);
      acc0 = __builtin_amdgcn_wmma_f32_16x16x4_f32(false, a, false, b0v,
                                                   (short)0, acc0, false, false);
      acc1 = __builtin_amdgcn_wmma_f32_16x16x4_f32(false, a, false, b1v,
                                                   (short)0, acc1, false, false);
    }
    __syncthreads();
  }

  // ---- epilogue + store: C/D layout M = r + 8*half, N = lr ----
#pragma unroll
  for (int r = 0; r < 8; ++r) {
    const int m = m0 + wm * 16 + r + 8 * half;
#pragma unroll
    for (int sub = 0; sub < 2; ++sub) {
      const int gn = n0 + wn * 32 + sub * 16 + lr;
      if (gn < N) {
        float val = (sub == 0) ? acc0[r] : acc1[r];
        if (MODE == 1) {
          val = tanhf(val + bias[gn]);
        } else if (MODE == 2) {
          float tv = tanhf(val + bias[gn]);
          val = (1.f - tv * tv) * aux[gn];
        } else if (MODE == 3) {
          float h = aux[(size_t)m * N + gn];
          val = val * (1.f - h * h);
        }
        OUT[(size_t)m * N + gn] = val;
      }
    }
  }
}

// ---------------------------------------------------------------------------
// Per-sample Poisson + friction combine. 64 threads = 1 sample.
// friction simplified: q = Bm v - BdotdE*(dE.v)/|dE|^2 ; out = C (C^T q)
// ---------------------------------------------------------------------------
__device__ __forceinline__ float breduce64(float v, float* red, int t) {
  __syncthreads();
  red[t] = v;
  __syncthreads();
#pragma unroll
  for (int off = 32; off > 0; off >>= 1) {
    if (t < off) red[t] += red[t + off];
    __syncthreads();
  }
  return red[0];
}

__global__ void __launch_bounds__(64)
combine_kernel(const float* __restrict__ dE, const float* __restrict__ dS,
               const float* __restrict__ Aflat, const float* __restrict__ Cm,
               const float* __restrict__ Bmat, float* __restrict__ out) {
  const int s = blockIdx.x;
  const int t = threadIdx.x;                 // row index 0..63
  __shared__ float sdE[64], sdS[64];
  __shared__ __align__(16) float sB[64 * 64];
  __shared__ __align__(16) float sC[64 * 16];
  __shared__ float sA[2016];
  __shared__ float red[64];
  __shared__ float sv[64], sq[64], sr[16];

  sdE[t] = dE[(size_t)s * 64 + t];
  sdS[t] = dS[(size_t)s * 64 + t];
  {
    const float4* Bp = (const float4*)(Bmat + (size_t)s * 4096);
    float4* db = (float4*)sB;
    for (int i = t; i < 1024; i += 64) db[i] = Bp[i];
    const float4* Cp = (const float4*)(Cm + (size_t)s * 1024);
    float4* dc = (float4*)sC;
    for (int i = t; i < 256; i += 64) dc[i] = Cp[i];
    const float* Ap = Aflat + (size_t)s * 2016;
    for (int i = t; i < 2016; i += 64) sA[i] = Ap[i];
  }
  __syncthreads();

  const float ei = sdE[t], si = sdS[t];
  const float dEsq = breduce64(ei * ei, red, t);
  const float dSsq = breduce64(si * si, red, t);
  const float dEdS = breduce64(ei * si, red, t);

  // antisymmetric A matvecs (tril(-1) flat index: i*(i-1)/2 + j)
  float AdE = 0.f, AdS = 0.f;
  for (int j = 0; j < t; ++j) {
    float a = sA[(t * (t - 1)) / 2 + j];
    AdE += a * sdE[j]; AdS += a * sdS[j];
  }
  for (int j = t + 1; j < 64; ++j) {
    float a = -sA[(j * (j - 1)) / 2 + t];
    AdE += a * sdE[j]; AdS += a * sdS[j];
  }

  // Bm matvecs
  float BdE = 0.f, BdS = 0.f;
  for (int j = 0; j < 64; ++j) {
    float b = sB[t * 64 + j];
    BdE += b * sdE[j]; BdS += b * sdS[j];
  }
  const float vi = BdS - BdE * (dEdS / dEsq);
  sv[t] = vi;
  const float dEv = breduce64(ei * vi, red, t);   // barrier also publishes sv

  float Bv = 0.f;
  for (int j = 0; j < 64; ++j) Bv += sB[t * 64 + j] * sv[j];
  sq[t] = Bv - BdE * (dEv / dEsq);
  __syncthreads();

  if (t < 16) {
    float rr = 0.f;
    for (int k = 0; k < 64; ++k) rr += sC[k * 16 + t] * sq[k];
    sr[t] = rr;
  }
  __syncthreads();

  float fric = 0.f;
#pragma unroll
  for (int c = 0; c < 16; ++c) fric += sC[t * 16 + c] * sr[c];

  const float dEAdS = breduce64(ei * AdS, red, t);
  const float pois = AdE + (dEAdS * si - dEdS * AdS) / dSsq;
  out[(size_t)s * 64 + t] = pois + fric;
}

// ---------------------------------------------------------------------------
extern "C" void kernel_launch(void* const* d_in, const int* in_sizes, int n_in,
                              void* d_out, int out_size, void* d_ws, size_t ws_size,
                              hipStream_t stream) {
  (void)in_sizes; (void)n_in; (void)out_size; (void)ws_size;
  const float* y = (const float*)d_in[1];
  // per-MLP weight pointers: W0,b0,W1,b1,W2 starting at index 2
  const float* eW0 = (const float*)d_in[2];  const float* eb0 = (const float*)d_in[3];
  const float* eW1 = (const float*)d_in[4];  const float* eb1 = (const float*)d_in[5];
  const float* eW2 = (const float*)d_in[6];
  const float* sW0 = (const float*)d_in[7];  const float* sb0 = (const float*)d_in[8];
  const float* sW1 = (const float*)d_in[9];  const float* sb1 = (const float*)d_in[10];
  const float* sW2 = (const float*)d_in[11];
  const float* pW0 = (const float*)d_in[12]; const float* pb0 = (const float*)d_in[13];
  const float* pW1 = (const float*)d_in[14]; const float* pb1 = (const float*)d_in[15];
  const float* pW2 = (const float*)d_in[16];
  const float* cW0 = (const float*)d_in[17]; const float* cb0 = (const float*)d_in[18];
  const float* cW1 = (const float*)d_in[19]; const float* cb1 = (const float*)d_in[20];
  const float* cW2 = (const float*)d_in[21];
  const float* bW0 = (const float*)d_in[22]; const float* bb0 = (const float*)d_in[23];
  const float* bW1 = (const float*)d_in[24]; const float* bb1 = (const float*)d_in[25];
  const float* bW2 = (const float*)d_in[26];

  float* ws = (float*)d_ws;
  const size_t BH = (size_t)NB * WID;
  float* H0  = ws;                       // [B,256]
  float* H1  = H0 + BH;                  // [B,256] (also V1)
  float* U0  = H1 + BH;                  // [B,256]
  float* dE  = U0 + BH;                  // [B,64]
  float* dS  = dE + (size_t)NB * SDIM;   // [B,64]
  float* Afl = dS + (size_t)NB * SDIM;   // [B,2016]
  float* Cm  = Afl + (size_t)NB * 2016;  // [B,1024]
  float* Bm  = Cm + (size_t)NB * 1024;   // [B,4096]

  dim3 blk(256);
  dim3 g256(NB / 64, WID / 64);          // N=256
  dim3 g64(NB / 64, 1);                  // N=64
  dim3 gA(NB / 64, (2016 + 63) / 64);    // N=2016
  dim3 gC(NB / 64, 1024 / 64);
  dim3 gB(NB / 64, 4096 / 64);

  // ---- energy gradient: dE ----
  gemm_f32_wmma<1, 1><<<g256, blk, 0, stream>>>(y,  eW0, SDIM, eb0, nullptr, H0, NB, WID, SDIM);
  gemm_f32_wmma<1, 2><<<g256, blk, 0, stream>>>(H0, eW1, WID,  eb1, eW2,    H1, NB, WID, WID);
  gemm_f32_wmma<0, 3><<<g256, blk, 0, stream>>>(H1, eW1, WID,  nullptr, H0, U0, NB, WID, WID);
  gemm_f32_wmma<0, 0><<<g64,  blk, 0, stream>>>(U0, eW0, SDIM, nullptr, nullptr, dE, NB, SDIM, WID);
  // ---- entropy gradient: dS ----
  gemm_f32_wmma<1, 1><<<g256, blk, 0, stream>>>(y,  sW0, SDIM, sb0, nullptr, H0, NB, WID, SDIM);
  gemm_f32_wmma<1, 2><<<g256, blk, 0, stream>>>(H0, sW1, WID,  sb1, sW2,    H1, NB, WID, WID);
  gemm_f32_wmma<0, 3><<<g256, blk, 0, stream>>>(H1, sW1, WID,  nullptr, H0, U0, NB, WID, WID);
  gemm_f32_wmma<0, 0><<<g64,  blk, 0, stream>>>(U0, sW0, SDIM, nullptr, nullptr, dS, NB, SDIM, WID);
  // ---- pA forward -> A_flat [B,2016] ----
  gemm_f32_wmma<1, 1><<<g256, blk, 0, stream>>>(y,  pW0, SDIM, pb0, nullptr, H0, NB, WID, SDIM);
  gemm_f32_wmma<1, 1><<<g256, blk, 0, stream>>>(H0, pW1, WID,  pb1, nullptr, H1, NB, WID, WID);
  gemm_f32_wmma<1, 0><<<gA,   blk, 0, stream>>>(H1, pW2, WID,  nullptr, nullptr, Afl, NB, 2016, WID);
  // ---- fC forward -> C [B,1024] ----
  gemm_f32_wmma<1, 1><<<g256, blk, 0, stream>>>(y,  cW0, SDIM, cb0, nullptr, H0, NB, WID, SDIM);
  gemm_f32_wmma<1, 1><<<g256, blk, 0, stream>>>(H0, cW1, WID,  cb1, nullptr, H1, NB, WID, WID);
  gemm_f32_wmma<1, 0><<<gC,   blk, 0, stream>>>(H1, cW2, WID,  nullptr, nullptr, Cm, NB, 1024, WID);
  // ---- fB forward -> Bm [B,4096] ----
  gemm_f32_wmma<1, 1><<<g256, blk, 0, stream>>>(y,  bW0, SDIM, bb0, nullptr, H0, NB, WID, SDIM);
  gemm_f32_wmma<1, 1><<<g256, blk, 0, stream>>>(H0, bW1, WID,  bb1, nullptr, H1, NB, WID, WID);
  gemm_f32_wmma<1, 0><<<gB,   blk, 0, stream>>>(H1, bW2, WID,  nullptr, nullptr, Bm, NB, 4096, WID);

  // ---- Poisson + friction combine ----
  combine_kernel<<<NB, 64, 0, stream>>>(dE, dS, Afl, Cm, Bm, (float*)d_out);
}